// CAPTOR_73701638800015
// MI455X (gfx1250) — compile-verified
//
#include <hip/hip_runtime.h>
#include <math.h>

#define NN 50000
#define EE 800000
#define HH 64
#define NTILES ((NN + 15) / 16)   // 3125, exact (3125*16 == 50000)
#define NBUCKETS 1024

typedef float v2f __attribute__((ext_vector_type(2)));
typedef float v8f __attribute__((ext_vector_type(8)));

__device__ __forceinline__ float waveReduceSum32(float v) {
#pragma unroll
    for (int off = 16; off > 0; off >>= 1) v += __shfl_xor(v, off, 32);
    return v;
}

// ---------------------------------------------------------------------------
// Per-layer init: m = -inf, denom = 0, crf = 0
// ---------------------------------------------------------------------------
__global__ __launch_bounds__(256) void init_node_kernel(float* __restrict__ m,
                                                        float* __restrict__ den,
                                                        float* __restrict__ crf) {
    int idx = blockIdx.x * 256 + threadIdx.x;
    if (idx < NN) { m[idx] = -INFINITY; den[idx] = 0.0f; }
    if (idx < NN * HH) crf[idx] = 0.0f;
}

// ---------------------------------------------------------------------------
// zs[i] = (W_fc @ h_i) . a_src ; zd[i] = (W_fc @ h_i) . a_dst
// One wave computes a 16-node x 64-col z strip via V_WMMA_F32_16X16X4_F32.
//
// B staging: P[p][c] (float2) = { W_fc[col][2p], W_fc[col][2p+1] } with
// col = (c - 16*(p&1)) mod 64.  The K-pair interleave makes each B operand a
// single contiguous 8B value (one ds_load_b64 straight into an even register
// pair, no assembly movs).  The 16-slot rotation on odd p makes the two lane
// halves (rows p=2kc and p=2kc+1) hit complementary 32-bank sets of the
// 64-bank LDS -> conflict-free b64 loads.
// ---------------------------------------------------------------------------
__global__ __launch_bounds__(256) void zsd_kernel(const float* __restrict__ h,
                                                  const float* __restrict__ W,
                                                  const float* __restrict__ attw,
                                                  float* __restrict__ zs,
                                                  float* __restrict__ zd) {
    __shared__ float2 P[32 * 64];   // 16 KB
    int tid = threadIdx.x;
    for (int i = tid; i < 32 * 64; i += 256) {
        int p = i >> 6, col = i & 63;
        int cs = (col + ((p & 1) << 4)) & 63;           // rotated slot
        P[(p << 6) + cs] = make_float2(W[col * HH + 2 * p], W[col * HH + 2 * p + 1]);
    }
    __syncthreads();

    int wave = tid >> 5;
    int lane = tid & 31;
    int tile = blockIdx.x * 8 + wave;
    if (tile >= NTILES) return;     // wave-uniform: EXEC stays all-ones for WMMA

    int base = tile * 16;
    int row  = lane & 15;           // A: M row / B: N col / D: N col
    int half = lane >> 4;           // lane half selects K sub-pair (koff = 2*half)

    // A row pointer with the K sub-pair folded in; per-kc advance = 16 B imm
    const float* hrow = h + (size_t)(base + row) * HH + 2 * half;

    // 4 per-column-group base pointers; row p = 2*kc + half -> +128 float2 per kc
    const float2* pb0 = &P[(half << 6) + (((0 + half) & 3) << 4) + row];
    const float2* pb1 = &P[(half << 6) + (((1 + half) & 3) << 4) + row];
    const float2* pb2 = &P[(half << 6) + (((2 + half) & 3) << 4) + row];
    const float2* pb3 = &P[(half << 6) + (((3 + half) & 3) << 4) + row];

    v8f c0 = {}, c1 = {}, c2 = {}, c3 = {};
#pragma unroll
    for (int kc = 0; kc < 16; ++kc) {
        v2f a  = *(const v2f*)(hrow + kc * 4);
        v2f b0 = *(const v2f*)(pb0 + kc * 128);
        v2f b1 = *(const v2f*)(pb1 + kc * 128);
        v2f b2 = *(const v2f*)(pb2 + kc * 128);
        v2f b3 = *(const v2f*)(pb3 + kc * 128);
        c0 = __builtin_amdgcn_wmma_f32_16x16x4_f32(false, a, false, b0, (short)0, c0, false, false);
        c1 = __builtin_amdgcn_wmma_f32_16x16x4_f32(false, a, false, b1, (short)0, c1, false, false);
        c2 = __builtin_amdgcn_wmma_f32_16x16x4_f32(false, a, false, b2, (short)0, c2, false, false);
        c3 = __builtin_amdgcn_wmma_f32_16x16x4_f32(false, a, false, b3, (short)0, c3, false, false);
    }

    // project z[m][:] onto a_src / a_dst; lane holds column (row + 16*g), g=0..3
    float as0 = attw[row],      as1 = attw[16 + row], as2 = attw[32 + row],  as3 = attw[48 + row];
    float ad0 = attw[64 + row], ad1 = attw[80 + row], ad2 = attw[96 + row],  ad3 = attw[112 + row];
    float ps[8], pd[8];
#pragma unroll
    for (int r = 0; r < 8; ++r) {
        ps[r] = c0[r] * as0 + c1[r] * as1 + c2[r] * as2 + c3[r] * as3;
        pd[r] = c0[r] * ad0 + c1[r] * ad1 + c2[r] * ad2 + c3[r] * ad3;
    }
    // reduce over the 16 lanes of each half (xor <= 8 stays inside the half)
#pragma unroll
    for (int off = 8; off > 0; off >>= 1) {
#pragma unroll
        for (int r = 0; r < 8; ++r) {
            ps[r] += __shfl_xor(ps[r], off, 32);
            pd[r] += __shfl_xor(pd[r], off, 32);
        }
    }
    if (row == 0) {                  // lanes 0 (rows 0..7) and 16 (rows 8..15)
        int rbase = base + half * 8;
#pragma unroll
        for (int r = 0; r < 8; ++r) { zs[rbase + r] = ps[r]; zd[rbase + r] = pd[r]; }
    }
}

// ---------------------------------------------------------------------------
// e = leakyrelu(zs[src] + zd[dst], 0.2) ; segment max into m[dst]
// ---------------------------------------------------------------------------
__global__ __launch_bounds__(256) void edge_logits_kernel(const float* __restrict__ zs,
                                                          const float* __restrict__ zd,
                                                          const int* __restrict__ src,
                                                          const int* __restrict__ dst,
                                                          float* __restrict__ e,
                                                          float* __restrict__ m) {
    int k = blockIdx.x * 256 + threadIdx.x;   // E divisible by 256
    float v = zs[src[k]] + zd[dst[k]];
    v = v > 0.0f ? v : 0.2f * v;
    e[k] = v;
    atomicMax(&m[dst[k]], v);                 // global_atomic_max_num_f32
}

// ---------------------------------------------------------------------------
// ex = exp(e - m[dst]) ; denom[dst] += ex
// ---------------------------------------------------------------------------
__global__ __launch_bounds__(256) void edge_ex_kernel(const float* __restrict__ e,
                                                      const float* __restrict__ m,
                                                      const int* __restrict__ dst,
                                                      float* __restrict__ ex,
                                                      float* __restrict__ den) {
    int k = blockIdx.x * 256 + threadIdx.x;
    int d = dst[k];
    float x = __expf(e[k] - m[d]);            // argument <= 0, no overflow
    ex[k] = x;
    atomicAdd(&den[d], x);
}

// ---------------------------------------------------------------------------
// crf[dst] += att * h[src]   (32 lanes per edge, float2 per lane)
// ---------------------------------------------------------------------------
__global__ __launch_bounds__(256) void aggregate_kernel(const float* __restrict__ ex,
                                                        const float* __restrict__ den,
                                                        const int* __restrict__ src,
                                                        const int* __restrict__ dst,
                                                        const float* __restrict__ hin,
                                                        float* __restrict__ crf) {
    int k = blockIdx.x * 8 + (threadIdx.x >> 5);   // E divisible by 8
    int lane = threadIdx.x & 31;
    int s = src[k], d = dst[k];
    float att = ex[k] / fmaxf(den[d], 1e-16f);
    const float2 hv = *(const float2*)(hin + (size_t)s * HH + lane * 2);
    float* out = crf + (size_t)d * HH + lane * 2;
    atomicAdd(out,     att * hv.x);
    atomicAdd(out + 1, att * hv.y);
}

// ---------------------------------------------------------------------------
// h = (ALPHA*emb + BETA*crf) / (ALPHA+BETA) = 0.5*(emb + crf)
// ---------------------------------------------------------------------------
__global__ __launch_bounds__(256) void update_kernel(const float* __restrict__ emb,
                                                     const float* __restrict__ crf,
                                                     float* __restrict__ hout) {
    int i = blockIdx.x * 256 + threadIdx.x;   // N*H divisible by 256
    hout[i] = 0.5f * (emb[i] + crf[i]);
}

// ---------------------------------------------------------------------------
// Loss machinery: bucketed double accumulation to avoid atomic serialization
// ---------------------------------------------------------------------------
__global__ void zero_buckets_kernel(double* __restrict__ b) {
    b[blockIdx.x * 256 + threadIdx.x] = 0.0;
}

__global__ __launch_bounds__(256) void loss_a_kernel(const float* __restrict__ h,
                                                     const float* __restrict__ emb,
                                                     double* __restrict__ buckets) {
    int n = blockIdx.x * 8 + (threadIdx.x >> 5);   // N divisible by 8
    int lane = threadIdx.x & 31;
    size_t o = (size_t)n * HH + lane * 2;
    float dx = h[o] - emb[o];
    float dy = h[o + 1] - emb[o + 1];
    float s = waveReduceSum32(dx * dx + dy * dy);
    if (lane == 0) atomicAdd(&buckets[blockIdx.x & (NBUCKETS - 1)], 0.5 * (double)s);
}

__global__ __launch_bounds__(256) void loss_b_kernel(const float* __restrict__ h,
                                                     const float* __restrict__ ex,
                                                     const float* __restrict__ den,
                                                     const int* __restrict__ src,
                                                     const int* __restrict__ dst,
                                                     double* __restrict__ buckets) {
    int k = blockIdx.x * 8 + (threadIdx.x >> 5);
    int lane = threadIdx.x & 31;
    int s = src[k], d = dst[k];
    size_t os = (size_t)s * HH + lane * 2;
    size_t od = (size_t)d * HH + lane * 2;
    float dx = h[od] - h[os];
    float dy = h[od + 1] - h[os + 1];
    float ss = waveReduceSum32(dx * dx + dy * dy);
    if (lane == 0) {
        float att = ex[k] / fmaxf(den[d], 1e-16f);
        atomicAdd(&buckets[blockIdx.x & (NBUCKETS - 1)],
                  0.5 * (double)(att * sqrtf(ss + 1e-12f)));
    }
}

__global__ __launch_bounds__(256) void finalize_kernel(const double* __restrict__ buckets,
                                                       float* __restrict__ out_loss) {
    __shared__ double sm[256];
    double s = 0.0;
    for (int i = threadIdx.x; i < NBUCKETS; i += 256) s += buckets[i];
    sm[threadIdx.x] = s;
    __syncthreads();
    for (int st = 128; st > 0; st >>= 1) {
        if (threadIdx.x < st) sm[threadIdx.x] += sm[threadIdx.x + st];
        __syncthreads();
    }
    if (threadIdx.x == 0) *out_loss = (float)(sm[0] / (double)NN);
}

// ---------------------------------------------------------------------------
extern "C" void kernel_launch(void* const* d_in, const int* in_sizes, int n_in,
                              void* d_out, int out_size, void* d_ws, size_t ws_size,
                              hipStream_t stream) {
    const float* emb  = (const float*)d_in[0];   // N*64
    const float* W    = (const float*)d_in[1];   // 64*64
    const float* attw = (const float*)d_in[2];   // 128
    const int*   src  = (const int*)d_in[3];     // E
    const int*   dst  = (const int*)d_in[4];     // E

    float* hout = (float*)d_out;                 // N*64
    float* loss = hout + (size_t)NN * HH;        // 1

    float*  wsf  = (float*)d_ws;
    float*  h1   = wsf;                          // N*64
    float*  crf  = h1 + (size_t)NN * HH;         // N*64
    float*  zs   = crf + (size_t)NN * HH;        // N
    float*  zd   = zs + NN;                      // N
    float*  mbuf = zd + NN;                      // N
    float*  den  = mbuf + NN;                    // N
    float*  ebuf = den + NN;                     // E
    float*  exb  = ebuf + EE;                    // E
    double* buckets = (double*)(exb + EE);       // NBUCKETS (offset is 8B-aligned)

    const float* hin = emb;
    for (int layer = 0; layer < 2; ++layer) {
        float* hdst = (layer == 0) ? h1 : hout;
        init_node_kernel<<<(NN * HH) / 256, 256, 0, stream>>>(mbuf, den, crf);
        zsd_kernel<<<(NTILES + 7) / 8, 256, 0, stream>>>(hin, W, attw, zs, zd);
        edge_logits_kernel<<<EE / 256, 256, 0, stream>>>(zs, zd, src, dst, ebuf, mbuf);
        edge_ex_kernel<<<EE / 256, 256, 0, stream>>>(ebuf, mbuf, dst, exb, den);
        aggregate_kernel<<<EE / 8, 256, 0, stream>>>(exb, den, src, dst, hin, crf);
        update_kernel<<<(NN * HH) / 256, 256, 0, stream>>>(emb, crf, hdst);
        hin = hdst;
    }
    zero_buckets_kernel<<<NBUCKETS / 256, 256, 0, stream>>>(buckets);
    loss_a_kernel<<<NN / 8, 256, 0, stream>>>(hout, emb, buckets);
    loss_b_kernel<<<EE / 8, 256, 0, stream>>>(hout, exb, den, src, dst, buckets);
    finalize_kernel<<<1, 256, 0, stream>>>(buckets, loss);
}